// NodeEncoder_3401614098589
// MI455X (gfx1250) — compile-verified
//
#include <hip/hip_runtime.h>

typedef float v2f __attribute__((ext_vector_type(2)));
typedef float v8f __attribute__((ext_vector_type(8)));

#define XS 65          // LDS stride for X tile (padded: avoids 16-way bank conflict)
#define WS 65          // LDS stride for W tiles

// ---------------------------------------------------------------------------
// Fused GEMM: T1 = X@W1, T2 = X@W2, ACC = X@Wr + br   (X: [N,64], W: [64,64])
// One block = 256 threads = 8 waves; each wave computes a 16-row strip via
// v_wmma_f32_16x16x4_f32 (fp32-exact matrix path, 16 WMMAs per 16x16 tile).
// ---------------------------------------------------------------------------
__global__ __launch_bounds__(256) void gemm3_wmma(
    const float* __restrict__ X,
    const float* __restrict__ W1, const float* __restrict__ W2,
    const float* __restrict__ Wr, const float* __restrict__ br,
    float* __restrict__ T1, float* __restrict__ T2, float* __restrict__ ACC,
    int N)
{
  __shared__ float sX[128 * XS];
  __shared__ float sW[3 * 64 * WS];
  __shared__ float sBr[64];

  const int tid = threadIdx.x;

  // cache the three 64x64 weight matrices in LDS (row stride padded to 65)
  for (int m = 0; m < 3; ++m) {
    const float* Wp = (m == 0) ? W1 : (m == 1) ? W2 : Wr;
    for (int idx = tid; idx < 64 * 16; idx += 256) {       // float4 granules
      int row = idx >> 4, c4 = idx & 15;
      float4 v = ((const float4*)Wp)[idx];
      float* p = &sW[m * 64 * WS + row * WS + c4 * 4];
      p[0] = v.x; p[1] = v.y; p[2] = v.z; p[3] = v.w;
    }
  }
  if (tid < 64) sBr[tid] = br[tid];

  // cache 128 rows of X in LDS
  {
    long rowBase = (long)blockIdx.x * 128;
    for (int idx = tid; idx < 128 * 16; idx += 256) {      // float4 granules
      int row = idx >> 4, c4 = idx & 15;
      long g = rowBase + row;
      float4 v = make_float4(0.f, 0.f, 0.f, 0.f);
      if (g < (long)N) v = ((const float4*)(X + g * 64))[c4];
      float* p = &sX[row * XS + c4 * 4];
      p[0] = v.x; p[1] = v.y; p[2] = v.z; p[3] = v.w;
    }
  }
  __syncthreads();

  const int wave = tid >> 5;
  const int lane = tid & 31;
  const int lm = lane & 15;      // A: row M / B,D: col N
  const int lh = lane >> 4;      // lane-half selector

  long row0 = ((long)blockIdx.x * 8 + wave) * 16;          // wave-uniform
  if (row0 >= (long)N) return;                             // EXEC stays all-1s

  const float* xrow = &sX[(wave * 16 + lm) * XS];

  for (int m = 0; m < 3; ++m) {
    const float* wm = &sW[m * 64 * WS];
    float* dstp = (m == 0) ? T1 : (m == 1) ? T2 : ACC;
    for (int n0 = 0; n0 < 64; n0 += 16) {
      v8f acc = {0.f, 0.f, 0.f, 0.f, 0.f, 0.f, 0.f, 0.f};
#pragma unroll
      for (int k0 = 0; k0 < 64; k0 += 4) {
        int kb = k0 + 2 * lh;                 // K = j + 2*(lane/16), j in {0,1}
        v2f a; a.x = xrow[kb];                a.y = xrow[kb + 1];
        v2f b; b.x = wm[kb * WS + n0 + lm];   b.y = wm[(kb + 1) * WS + n0 + lm];
        acc = __builtin_amdgcn_wmma_f32_16x16x4_f32(
            false, a, false, b, (short)0, acc, false, false);
      }
      float add = (m == 2) ? sBr[n0 + lm] : 0.0f;
#pragma unroll
      for (int v = 0; v < 8; ++v) {           // D[v + 8*lh][lm]
        long gr = row0 + v + 8 * lh;
        dstp[gr * 64 + n0 + lm] = acc[v] + add;
      }
    }
  }
}

// ---------------------------------------------------------------------------
// Degree kernels (degrees are identical for both layers -> computed once)
// ---------------------------------------------------------------------------
__global__ void zero2(float* __restrict__ a, float* __restrict__ b, int n) {
  int i = blockIdx.x * blockDim.x + threadIdx.x;
  if (i < n) { a[i] = 0.f; b[i] = 0.f; }
}

__global__ void degree_count(const int* __restrict__ src, const int* __restrict__ dst,
                             float* __restrict__ dS, float* __restrict__ dD, int E) {
  int i = blockIdx.x * blockDim.x + threadIdx.x;
  if (i < E) {
    atomicAdd(&dD[dst[i]], 1.0f);
    atomicAdd(&dS[src[i]], 1.0f);
  }
}

__global__ void degree_invert(float* __restrict__ dS, float* __restrict__ dD, int n) {
  int i = blockIdx.x * blockDim.x + threadIdx.x;
  if (i < n) {
    dS[i] = 1.0f / fmaxf(dS[i], 1.0f);
    dD[i] = 1.0f / fmaxf(dD[i], 1.0f);
  }
}

// ---------------------------------------------------------------------------
// Edge scatter-mean: one wave per edge, 2 floats/lane (coalesced 256B rows),
// pre-scaled by inverse degree, atomically accumulated (L2-resident: 25.6MB).
// ---------------------------------------------------------------------------
__global__ __launch_bounds__(256) void edge_scatter(
    const float* __restrict__ T1, const float* __restrict__ T2,
    const int* __restrict__ src, const int* __restrict__ dst,
    const float* __restrict__ invS, const float* __restrict__ invD,
    float* __restrict__ acc, int E)
{
  int wid  = (blockIdx.x * blockDim.x + threadIdx.x) >> 5;
  int lane = threadIdx.x & 31;
  int nw   = (gridDim.x * blockDim.x) >> 5;
  for (int e = wid; e < E; e += nw) {
    int s = src[e], d = dst[e];
    float wd = invD[d];
    float ws = invS[s];
    float2 a = ((const float2*)(T1 + (long)s * 64))[lane];  // lin1(x)[src]
    float2 b = ((const float2*)(T2 + (long)d * 64))[lane];  // lin2(x)[dst]
    float* pd = acc + (long)d * 64 + lane * 2;
    atomicAdd(pd,     a.x * wd);
    atomicAdd(pd + 1, a.y * wd);
    float* ps = acc + (long)s * 64 + lane * 2;
    atomicAdd(ps,     b.x * ws);
    atomicAdd(ps + 1, b.y * ws);
  }
}

__global__ void relu_inplace(float* __restrict__ p, long n) {
  long i = (long)blockIdx.x * blockDim.x + threadIdx.x;
  long stride = (long)gridDim.x * blockDim.x;
  for (; i < n; i += stride) p[i] = fmaxf(p[i], 0.0f);
}

// ---------------------------------------------------------------------------
extern "C" void kernel_launch(void* const* d_in, const int* in_sizes, int n_in,
                              void* d_out, int out_size, void* d_ws, size_t ws_size,
                              hipStream_t stream)
{
  const float* x    = (const float*)d_in[0];
  const int*   ei   = (const int*)  d_in[1];
  const float* W1_0 = (const float*)d_in[2];
  const float* W2_0 = (const float*)d_in[3];
  const float* Wr_0 = (const float*)d_in[4];
  const float* br_0 = (const float*)d_in[5];
  const float* W1_1 = (const float*)d_in[6];
  const float* W2_1 = (const float*)d_in[7];
  const float* Wr_1 = (const float*)d_in[8];
  const float* br_1 = (const float*)d_in[9];

  const int N = in_sizes[0] / 64;
  const int E = in_sizes[1] / 2;
  const int* src = ei;          // edge_index[0]
  const int* dst = ei + E;      // edge_index[1]

  float* ws   = (float*)d_ws;
  size_t nd   = (size_t)N * 64;
  float* T1   = ws;             // lin1 output
  float* T2   = T1 + nd;        // lin2 output
  float* H    = T2 + nd;        // layer-1 accumulator / hidden activation
  float* invD = H + nd;         // 1/max(deg_dst,1)
  float* invS = invD + N;       // 1/max(deg_src,1)
  float* out  = (float*)d_out;

  const int tb = 256;

  // degrees (shared by both layers)
  zero2        <<<(N + tb - 1) / tb, tb, 0, stream>>>(invS, invD, N);
  degree_count <<<(E + tb - 1) / tb, tb, 0, stream>>>(src, dst, invS, invD, E);
  degree_invert<<<(N + tb - 1) / tb, tb, 0, stream>>>(invS, invD, N);

  const int tiles   = (N + 15) / 16;
  const int gblocks = (tiles + 7) / 8;

  // ---- layer 1: H = relu(x@Wr0 + br0 + mean_dst(x@W1_0[src]) + mean_src(x@W2_0[dst]))
  gemm3_wmma  <<<gblocks, 256, 0, stream>>>(x, W1_0, W2_0, Wr_0, br_0, T1, T2, H, N);
  edge_scatter<<<2048,    256, 0, stream>>>(T1, T2, src, dst, invS, invD, H, E);
  relu_inplace<<<4096,    256, 0, stream>>>(H, (long)nd);

  // ---- layer 2: out = relu(H@Wr1 + br1 + mean_dst(H@W1_1[src]) + mean_src(H@W2_1[dst]))
  gemm3_wmma  <<<gblocks, 256, 0, stream>>>(H, W1_1, W2_1, Wr_1, br_1, T1, T2, out, N);
  edge_scatter<<<2048,    256, 0, stream>>>(T1, T2, src, dst, invS, invD, out, E);
  relu_inplace<<<4096,    256, 0, stream>>>(out, (long)nd);
}